// MyPreprocessingLayer_17154099380519
// MI455X (gfx1250) — compile-verified
//
#include <hip/hip_runtime.h>

typedef __attribute__((ext_vector_type(2))) float v2f;
typedef __attribute__((ext_vector_type(8))) float v8f;

#define NROWS 8192
#define NCOLS 541
#define NFEAT 512
#define NLOC  29

// One block (8 wave32s) per input row. Each wave computes four 16-feature
// M-tiles of the 512x29 outer product as rank-1 updates via
// V_WMMA_F32_16X16X4_F32 (K=0 carries the data, K=1..3 zeroed).
// Result is bit-identical to a scalar f32 multiply (single RNE-rounded FMA,
// C = 0). The kernel is HBM-store-bound (~486 MB out => ~21.6 us floor).
__global__ __launch_bounds__(256)
void outer_wmma_kernel(const float* __restrict__ in, float* __restrict__ out)
{
    const int row  = blockIdx.x;
    const int tid  = threadIdx.x;
    const int lane = tid & 31;
    const int wave = tid >> 5;              // 0..7

    const float* rowp = in + (size_t)row * NCOLS;
    const float* feat = rowp;               // first 512 columns
    const float* loc  = rowp + NFEAT;       // last 29 columns

    // B fragments (4x16, K rows): only K=0 (VGPR0, lanes 0..15) is non-zero.
    v2f b0; b0.x = (lane < 16) ? loc[lane]      : 0.0f; b0.y = 0.0f;
    v2f b1; b1.x = (lane < 13) ? loc[16 + lane] : 0.0f; b1.y = 0.0f;

    float* orow = out + (size_t)row * (NFEAT * NLOC);

    const int hi = lane >> 4;               // D tile: lanes 16..31 hold M+8
    const int n  = lane & 15;               // D tile: N = lane % 16

#pragma unroll
    for (int t = 0; t < 4; ++t) {
        const int mbase = (wave * 4 + t) * 16;

        // A fragment (16x4): only K=0 (VGPR0, lanes 0..15) is non-zero.
        v2f a;
        a.x = (lane < 16) ? feat[mbase + lane] : 0.0f;
        a.y = 0.0f;

        v8f c = {};
        // 8 args: (neg_a, A, neg_b, B, c_mod, C, reuse_a, reuse_b)
        v8f d0 = __builtin_amdgcn_wmma_f32_16x16x4_f32(
            false, a, false, b0, (short)0, c, false, false);
        v8f d1 = __builtin_amdgcn_wmma_f32_16x16x4_f32(
            false, a, false, b1, (short)0, c, false, false);

        // Element (M = mbase + v + 8*hi, N = n) -> orow[M*29 + N]
        float* p = orow + (size_t)(mbase + hi * 8) * NLOC + n;
#pragma unroll
        for (int v = 0; v < 8; ++v)
            p[v * NLOC] = d0[v];

        if (n < 13) {                        // second N-tile covers cols 16..28
            float* q = p + 16;
#pragma unroll
            for (int v = 0; v < 8; ++v)
                q[v * NLOC] = d1[v];
        }
    }
}

extern "C" void kernel_launch(void* const* d_in, const int* in_sizes, int n_in,
                              void* d_out, int out_size, void* d_ws, size_t ws_size,
                              hipStream_t stream) {
    (void)in_sizes; (void)n_in; (void)out_size; (void)d_ws; (void)ws_size;
    const float* in = (const float*)d_in[0];
    float* out = (float*)d_out;
    outer_wmma_kernel<<<NROWS, 256, 0, stream>>>(in, out);
}